// MAC_19731079758157
// MI455X (gfx1250) — compile-verified
//
#include <hip/hip_runtime.h>
#include <hip/hip_bf16.h>
#include <math.h>

#define HID   256
#define COMP  32
#define T_TOK 8
#define NVOC  257   // vocab rows incl. EOS
#define ROWS  16    // rows per block (= 2 agent groups of 8)

typedef __attribute__((ext_vector_type(2))) float v2f;
typedef __attribute__((ext_vector_type(8))) float v8f;

// ---- fp32 WMMA helpers (V_WMMA_F32_16X16X4_F32) ----------------------------
static __device__ __forceinline__ v8f wmma4(v2f a, v2f b, v8f c) {
    return __builtin_amdgcn_wmma_f32_16x16x4_f32(false, a, false, b, (short)0, c,
                                                 false, false);
}

// A-fragment (16x4) from an LDS matrix [16][HID], k-step base k0.
static __device__ __forceinline__ v2f ldsA(const float* A, int k0, int lane) {
    int row = lane & 15;
    int kk  = k0 + ((lane >> 4) << 1);
    v2f a; a.x = A[row * HID + kk]; a.y = A[row * HID + kk + 1];
    return a;
}

// B-fragment (4x16) from a global row-major [K][ldb] weight, tile col base n0.
static __device__ __forceinline__ v2f gloB(const float* W, int ldb, int n0,
                                           int k0, int lane) {
    int n  = n0 + (lane & 15);
    int kk = k0 + ((lane >> 4) << 1);
    v2f b; b.x = W[kk * ldb + n]; b.y = W[(kk + 1) * ldb + n];
    return b;
}

// ---- main fused kernel: Q/V GEMMs + 8-step recurrence + agent group-sum ----
__global__ __launch_bounds__(256)
void comm_main_kernel(const float* __restrict__ hid,
                      const float* __restrict__ eps,
                      const float* __restrict__ W_Q,  const float* __restrict__ b_Q,
                      const float* __restrict__ W_K,  const float* __restrict__ b_K,
                      const float* __restrict__ W_mu, const float* __restrict__ b_mu,
                      const float* __restrict__ W_var,const float* __restrict__ b_var,
                      const float* __restrict__ vocab,
                      float* __restrict__ comm_ws, int B)
{
    extern __shared__ float smem[];
    float* sMsg  = smem;                    // [16][256] hidden, then message
    float* sQ    = smem + 4096;             // [16][256]
    float* sV    = smem + 8192;             // [16][256]
    float* sK    = smem + 12288;            // [16][256] K, then attn*V
    float* sVoc  = smem + 16384;            // [257][32]
    float* sMu   = smem + 16384 + NVOC*COMP;        // [16][32]
    float* sLv   = sMu  + ROWS*COMP;                // [16][32]
    float* sTok  = sLv  + ROWS*COMP;                // [16][32]
    float* sMask = sTok + ROWS*COMP;                // [16]
    float* sFin  = sMask + ROWS;                    // [16]

    const int tid  = threadIdx.x;
    const int wv   = tid >> 5;
    const int lane = tid & 31;
    const int row0 = blockIdx.x * ROWS;
    const float NF = 0.0625f;   // 1/sqrt(256)

    // ---- async-DMA the 16x256 hidden tile into LDS (GVS mode, B128/lane) ---
    {
        const float* gbase = hid + (size_t)row0 * HID;
        for (int e = tid * 4; e < ROWS * HID; e += 1024) {
            unsigned lds_addr = (unsigned)(size_t)(sMsg + e);   // low 32b = LDS ofs
            unsigned goff     = (unsigned)e * 4u;               // byte offset
            asm volatile("global_load_async_to_lds_b128 %0, %1, %2"
                         :: "v"(lds_addr), "v"(goff), "s"(gbase)
                         : "memory");
        }
    }
    for (int e = tid; e < NVOC * COMP; e += 256) sVoc[e] = vocab[e];
    if (tid < ROWS) { sMask[tid] = 1.0f; sFin[tid] = 0.0f; }
    asm volatile("s_wait_asynccnt 0" ::: "memory");
    __syncthreads();

    // ---- Q = (h@W_Q + bQ)*NF ; V = h@W_K + bK  (wave wv owns cols wv*32..+31)
    {
        const int n0 = wv * 32;
        v8f q0 = {}, q1 = {}, v0 = {}, v1 = {};
        for (int k0 = 0; k0 < HID; k0 += 4) {
            v2f a = ldsA(sMsg, k0, lane);
            q0 = wmma4(a, gloB(W_Q, HID, n0,      k0, lane), q0);
            q1 = wmma4(a, gloB(W_Q, HID, n0 + 16, k0, lane), q1);
            v0 = wmma4(a, gloB(W_K, HID, n0,      k0, lane), v0);
            v1 = wmma4(a, gloB(W_K, HID, n0 + 16, k0, lane), v1);
        }
        int col0 = n0 + (lane & 15), col1 = col0 + 16;
        int rb = (lane >> 4) << 3;
        float bq0 = b_Q[col0], bq1 = b_Q[col1], bk0 = b_K[col0], bk1 = b_K[col1];
        #pragma unroll
        for (int r = 0; r < 8; ++r) {
            sQ[(rb + r) * HID + col0] = (q0[r] + bq0) * NF;
            sQ[(rb + r) * HID + col1] = (q1[r] + bq1) * NF;
            sV[(rb + r) * HID + col0] =  v0[r] + bk0;
            sV[(rb + r) * HID + col1] =  v1[r] + bk1;
        }
    }
    __syncthreads();
    for (int e = tid; e < ROWS * HID; e += 256) sMsg[e] = 0.0f;   // message = 0

    // ---- sequential token recurrence -------------------------------------
    for (int it = 0; it < T_TOK; ++it) {
        __syncthreads();
        // K = (msg @ W_K + bK) * NF.  Only the first it*32 columns of msg are
        // nonzero (tokens are appended sequentially), so truncate the K-loop.
        {
            const int n0 = wv * 32;
            const int kmax = it * COMP;          // 0, 32, ..., 224
            v8f c0 = {}, c1 = {};
            for (int k0 = 0; k0 < kmax; k0 += 4) {
                v2f a = ldsA(sMsg, k0, lane);
                c0 = wmma4(a, gloB(W_K, HID, n0,      k0, lane), c0);
                c1 = wmma4(a, gloB(W_K, HID, n0 + 16, k0, lane), c1);
            }
            int col0 = n0 + (lane & 15), col1 = col0 + 16;
            int rb = (lane >> 4) << 3;
            float bk0 = b_K[col0], bk1 = b_K[col1];
            #pragma unroll
            for (int r = 0; r < 8; ++r) {
                sK[(rb + r) * HID + col0] = (c0[r] + bk0) * NF;
                sK[(rb + r) * HID + col1] = (c1[r] + bk1) * NF;
            }
        }
        __syncthreads();
        // softmin(Q*K) * V  (wave wv owns rows 2wv, 2wv+1; wave32 shuffles)
        #pragma unroll
        for (int rr = 0; rr < 2; ++rr) {
            int r = wv * 2 + rr;
            float vals[8]; float mx = -3.0e38f;
            #pragma unroll
            for (int j = 0; j < 8; ++j) {
                int c = lane + j * 32;
                float l = -(sQ[r * HID + c] * sK[r * HID + c]);
                vals[j] = l; mx = fmaxf(mx, l);
            }
            #pragma unroll
            for (int off = 16; off > 0; off >>= 1)
                mx = fmaxf(mx, __shfl_xor(mx, off, 32));
            float s = 0.0f;
            #pragma unroll
            for (int j = 0; j < 8; ++j) { vals[j] = __expf(vals[j] - mx); s += vals[j]; }
            #pragma unroll
            for (int off = 16; off > 0; off >>= 1) s += __shfl_xor(s, off, 32);
            float inv = 1.0f / s;
            #pragma unroll
            for (int j = 0; j < 8; ++j) {
                int c = lane + j * 32;
                sK[r * HID + c] = vals[j] * inv * sV[r * HID + c];  // hidden_attn
            }
        }
        __syncthreads();
        // mu / log_var: [16,256]x[256,32]; waves 0..3 take one 16x16 tile each
        if (wv < 4) {
            const float* Wx   = (wv < 2) ? W_mu : W_var;
            const float* bx   = (wv < 2) ? b_mu : b_var;
            float*       dst  = (wv < 2) ? sMu  : sLv;
            const int n0 = (wv & 1) * 16;
            v8f c0 = {};
            for (int k0 = 0; k0 < HID; k0 += 4) {
                v2f a = ldsA(sK, k0, lane);
                c0 = wmma4(a, gloB(Wx, COMP, n0, k0, lane), c0);
            }
            int col = n0 + (lane & 15);
            int rb = (lane >> 4) << 3;
            float bb = bx[col];
            #pragma unroll
            for (int r = 0; r < 8; ++r) dst[(rb + r) * COMP + col] = c0[r] + bb;
        }
        __syncthreads();
        // token = eps*exp(0.5 lv)+mu ; then per-dim min over 257 vocab rows
        for (int e = tid; e < ROWS * COMP; e += 256) {
            int r = e >> 5, c = e & 31;
            float t = eps[((size_t)it * B + row0 + r) * COMP + c]
                      * __expf(0.5f * sLv[e]) + sMu[e];
            float best = 3.0e38f;
            for (int v = 0; v < NVOC; ++v) {
                float d = t - sVoc[v * COMP + c];
                best = fminf(best, d * d);
            }
            sTok[e] = best;
        }
        __syncthreads();
        // mask / EOS / duplicate logic: one lane per row
        if (tid < ROWS) {
            int r = tid;
            float msk = sMask[r];
            float tl[COMP];
            #pragma unroll
            for (int c = 0; c < COMP; ++c) tl[c] = sTok[r * COMP + c] * msk;
            float de = 0.0f;
            #pragma unroll
            for (int c = 0; c < COMP; ++c) {
                float d = tl[c] - sVoc[256 * COMP + c]; de += d * d;
            }
            bool hit = de < 1e-6f;
            for (int j = 0; j < it; ++j) {
                float dp = 0.0f;
                for (int c = 0; c < COMP; ++c) {
                    float d = tl[c] - sMsg[r * HID + j * COMP + c]; dp += d * d;
                }
                if (dp < 1e-6f) hit = true;
            }
            bool fin = (sFin[r] != 0.0f);
            bool neweos = hit && !fin;
            if (hit) { sMask[r] = 0.0f; sFin[r] = 1.0f; }
            for (int c = 0; c < COMP; ++c)
                sMsg[r * HID + it * COMP + c] = neweos ? sVoc[256 * COMP + c] : tl[c];
        }
    }
    __syncthreads();
    // agent group-sum (/7): block holds 2 groups of 8 rows; emit unique rows
    for (int e = tid; e < 2 * HID; e += 256) {
        int g = e >> 8, c = e & 255;
        float s = 0.0f;
        #pragma unroll
        for (int a = 0; a < 8; ++a) s += sMsg[(g * 8 + a) * HID + c];
        comm_ws[((size_t)blockIdx.x * 2 + g) * HID + c] = s * (1.0f / 7.0f);
    }
}

// ---- tail: out = relu((cs@W_sum+b)@W_head+b) on B/8 unique rows, fan out 8x
__global__ __launch_bounds__(256)
void comm_head_kernel(const float* __restrict__ comm_ws,
                      const float* __restrict__ W_sum,  const float* __restrict__ b_sum,
                      const float* __restrict__ W_head, const float* __restrict__ b_head,
                      float* __restrict__ out)
{
    __shared__ float sA[ROWS * HID];
    __shared__ float sT[ROWS * HID];
    const int tid = threadIdx.x;
    const int wv = tid >> 5, lane = tid & 31;
    const int u0 = blockIdx.x * ROWS;

    // async-DMA the 16x256 comm_sum tile into LDS
    {
        const float* gbase = comm_ws + (size_t)u0 * HID;
        for (int e = tid * 4; e < ROWS * HID; e += 1024) {
            unsigned lds_addr = (unsigned)(size_t)(sA + e);
            unsigned goff     = (unsigned)e * 4u;
            asm volatile("global_load_async_to_lds_b128 %0, %1, %2"
                         :: "v"(lds_addr), "v"(goff), "s"(gbase)
                         : "memory");
        }
    }
    asm volatile("s_wait_asynccnt 0" ::: "memory");
    __syncthreads();
    {
        const int n0 = wv * 32;
        v8f c0 = {}, c1 = {};
        for (int k0 = 0; k0 < HID; k0 += 4) {
            v2f a = ldsA(sA, k0, lane);
            c0 = wmma4(a, gloB(W_sum, HID, n0,      k0, lane), c0);
            c1 = wmma4(a, gloB(W_sum, HID, n0 + 16, k0, lane), c1);
        }
        int col0 = n0 + (lane & 15), col1 = col0 + 16;
        int rb = (lane >> 4) << 3;
        float bb0 = b_sum[col0], bb1 = b_sum[col1];
        #pragma unroll
        for (int r = 0; r < 8; ++r) {
            sT[(rb + r) * HID + col0] = c0[r] + bb0;
            sT[(rb + r) * HID + col1] = c1[r] + bb1;
        }
    }
    __syncthreads();
    {
        const int n0 = wv * 32;
        v8f c0 = {}, c1 = {};
        for (int k0 = 0; k0 < HID; k0 += 4) {
            v2f a = ldsA(sT, k0, lane);
            c0 = wmma4(a, gloB(W_head, HID, n0,      k0, lane), c0);
            c1 = wmma4(a, gloB(W_head, HID, n0 + 16, k0, lane), c1);
        }
        int col0 = n0 + (lane & 15), col1 = col0 + 16;
        int rb = (lane >> 4) << 3;
        float bb0 = b_head[col0], bb1 = b_head[col1];
        #pragma unroll
        for (int r = 0; r < 8; ++r) {
            float o0 = fmaxf(c0[r] + bb0, 0.0f);
            float o1 = fmaxf(c1[r] + bb1, 0.0f);
            size_t u = (size_t)(u0 + rb + r);
            #pragma unroll
            for (int a = 0; a < 8; ++a) {          // broadcast to 8 agents
                out[(u * 8 + a) * HID + col0] = o0;
                out[(u * 8 + a) * HID + col1] = o1;
            }
        }
    }
}

extern "C" void kernel_launch(void* const* d_in, const int* in_sizes, int n_in,
                              void* d_out, int out_size, void* d_ws, size_t ws_size,
                              hipStream_t stream) {
    const float* hid    = (const float*)d_in[0];
    const float* eps    = (const float*)d_in[1];
    const float* W_Q    = (const float*)d_in[2];
    const float* b_Q    = (const float*)d_in[3];
    const float* W_K    = (const float*)d_in[4];
    const float* b_K    = (const float*)d_in[5];
    const float* W_mu   = (const float*)d_in[6];
    const float* b_mu   = (const float*)d_in[7];
    const float* W_var  = (const float*)d_in[8];
    const float* b_var  = (const float*)d_in[9];
    const float* vocab  = (const float*)d_in[10];
    const float* W_sum  = (const float*)d_in[11];
    const float* b_sum  = (const float*)d_in[12];
    const float* W_head = (const float*)d_in[13];
    const float* b_head = (const float*)d_in[14];
    float* out = (float*)d_out;
    float* ws  = (float*)d_ws;   // [B/8, 256] unique comm_sum rows

    const int B = in_sizes[0] / HID;
    const size_t smem_bytes =
        (4 * ROWS * HID + NVOC * COMP + 3 * ROWS * COMP + 2 * ROWS) * sizeof(float);

    comm_main_kernel<<<B / ROWS, 256, smem_bytes, stream>>>(
        hid, eps, W_Q, b_Q, W_K, b_K, W_mu, b_mu, W_var, b_var, vocab, ws, B);
    comm_head_kernel<<<(B / 8) / ROWS, 256, 0, stream>>>(
        ws, W_sum, b_sum, W_head, b_head, out);
}